// Classifier_23441931501808
// MI455X (gfx1250) — compile-verified
//
#include <hip/hip_runtime.h>
#include <hip/hip_bf16.h>

typedef float v2f __attribute__((ext_vector_type(2)));
typedef float v4f __attribute__((ext_vector_type(4)));
typedef float v8f __attribute__((ext_vector_type(8)));

// ---------------------------------------------------------------------------
// pred[b,k] = sum_{n<256} x[b, k*256 + n]
// Viewed as rows of a [B*K, 256] row-major matrix (row r -> pred flat idx r).
//
// One wave32 handles 16 rows via V_WMMA_F32_16X16X4_F32:
//   A (16x4 fp32)  = 16 rows x 4 consecutive columns of x
//   B (4x16 fp32)  = all ones  ->  every column of D = running row sums
//   C/D (16x16)    = fp32 accumulator, 64 WMMA steps cover 256 columns
//
// A-matrix VGPR layout (ISA 7.12.2, 32-bit A 16x4):
//   lanes 0-15 : VGPR0=A[m][0], VGPR1=A[m][1]   (m = lane)
//   lanes 16-31: VGPR0=A[m][2], VGPR1=A[m][3]   (m = lane-16)
// -> each lane issues one b64 load per step.
//
// D layout (ISA 7.12.2, 32-bit C/D 16x16):
//   lanes 0-15 : VGPR g = D[g][lane]
//   lanes 16-31: VGPR g = D[8+g][lane-16]
// -> lane 0 holds rows 0..7 (col 0), lane 16 holds rows 8..15 (col 0).
// ---------------------------------------------------------------------------
__global__ __launch_bounds__(256) void segsum_wmma_kernel(
    const float* __restrict__ x, float* __restrict__ pred) {
  const int lane = threadIdx.x & 31;
  const int waveInBlock = threadIdx.x >> 5;
  const long wave = (long)blockIdx.x * 8 + waveInBlock;   // 8 waves / block
  const long rowBase = wave * 16;                          // 16 rows / wave

  const int m = lane & 15;            // matrix row within the 16-row tile
  const int koff = (lane >> 4) << 1;  // 0 for lanes 0-15, 2 for lanes 16-31

  const float* rowPtr = x + (rowBase + (long)m) * 256 + koff;

  v8f acc = {};                       // C/D accumulator (16x16 fp32 tile)
  v2f ones; ones[0] = 1.0f; ones[1] = 1.0f;   // B = all ones

#pragma unroll 8
  for (int s = 0; s < 64; ++s) {
    v2f a = *(const v2f*)(rowPtr + s * 4);    // A fragment: 2 fp32 per lane
    // 8 args: (neg_a, A, neg_b, B, c_mod, C, reuse_a, reuse_b)
    acc = __builtin_amdgcn_wmma_f32_16x16x4_f32(
        false, a, false, ones, (short)0, acc, false, false);
  }

  // Column 0 of D carries the 16 row sums. Lanes 0 and 16 write 8 each.
  if ((lane & 15) == 0) {
    float* p = pred + rowBase + ((lane >> 4) << 3);
    v4f lo; lo[0] = acc[0]; lo[1] = acc[1]; lo[2] = acc[2]; lo[3] = acc[3];
    v4f hi; hi[0] = acc[4]; hi[1] = acc[5]; hi[2] = acc[6]; hi[3] = acc[7];
    *(v4f*)(p + 0) = lo;
    *(v4f*)(p + 4) = hi;
  }
}

// ---------------------------------------------------------------------------
// y = repeat_interleave(eye(64), 256, axis=0) -> [16384, 64] one-hot rows.
// y[r][k] = (r >> 8 == k). Written fresh every call (harness poisons d_out).
// One float4 per thread; k-groups of 4 stay within one row (64 % 4 == 0).
// ---------------------------------------------------------------------------
__global__ __launch_bounds__(256) void y_fill_kernel(float* __restrict__ y) {
  const int i = blockIdx.x * blockDim.x + threadIdx.x;  // float4 index
  const int r = i >> 4;                                  // 16 float4s per row
  const int k0 = (i & 15) << 2;                          // first k of group
  const int hot = r >> 8;                                // the one-hot column
  v4f v;
  v[0] = (k0 + 0 == hot) ? 1.0f : 0.0f;
  v[1] = (k0 + 1 == hot) ? 1.0f : 0.0f;
  v[2] = (k0 + 2 == hot) ? 1.0f : 0.0f;
  v[3] = (k0 + 3 == hot) ? 1.0f : 0.0f;
  *(v4f*)(y + ((long)i << 2)) = v;
}

extern "C" void kernel_launch(void* const* d_in, const int* in_sizes, int n_in,
                              void* d_out, int out_size, void* d_ws, size_t ws_size,
                              hipStream_t stream) {
  (void)in_sizes; (void)n_in; (void)d_ws; (void)ws_size; (void)out_size;

  const float* x = (const float*)d_in[0];
  float* pred = (float*)d_out;                 // 4096*64 = 262144 floats
  float* y    = (float*)d_out + 4096 * 64;     // 16384*64 = 1048576 floats

  // pred: 262144 rows / 16 rows-per-wave = 16384 waves; 8 waves per block.
  segsum_wmma_kernel<<<2048, 256, 0, stream>>>(x, pred);

  // y: 1048576 floats / 4 per thread = 262144 threads.
  y_fill_kernel<<<1024, 256, 0, stream>>>(y);
}